// WeightedFilter_59012850647116
// MI455X (gfx1250) — compile-verified
//
#include <hip/hip_runtime.h>

// Problem constants (from reference setup_inputs / init_kwargs)
#define KSZ 7
#define RAD 3
#define BB 4
#define CC 16
#define HH 720
#define WW 1280

// Tiling
#define TILE_X 32
#define TILE_Y 16
#define NTHR   512
#define YR (TILE_Y + 2 * RAD)   // 22 rows incl. halo
#define XR (TILE_X + 2 * RAD)   // 38 cols incl. halo
#define XP 40                   // padded LDS pitch (float4 units, 16B aligned rows)

typedef float v4f __attribute__((ext_vector_type(4)));

typedef __attribute__((address_space(1))) int glb_i32_t;
typedef __attribute__((address_space(3))) int lds_i32_t;

// Async global->LDS copy (CDNA5 GLOBAL_LOAD_ASYNC_TO_LDS_B32, tracked by ASYNCcnt)
__device__ __forceinline__ void async_g2l_b32(const float* g, float* l) {
#if __has_builtin(__builtin_amdgcn_global_load_async_to_lds_b32)
  __builtin_amdgcn_global_load_async_to_lds_b32(
      (glb_i32_t*)g, (lds_i32_t*)l, 0, 0);
#else
  lds_i32_t* lp = (lds_i32_t*)l;
  glb_i32_t* gp = (glb_i32_t*)g;
  unsigned loff = (unsigned)(unsigned long long)lp;
  asm volatile("global_load_async_to_lds_b32 %0, %1, off"
               :: "v"(loff), "v"(gp) : "memory");
#endif
}

__device__ __forceinline__ void wait_async0() {
#if __has_builtin(__builtin_amdgcn_s_wait_asynccnt)
  __builtin_amdgcn_s_wait_asynccnt(0);
#else
  asm volatile("s_wait_asynccnt 0" ::: "memory");
#endif
}

__global__ __launch_bounds__(NTHR) void wfilter7x7_kernel(
    const float* __restrict__ x, const float* __restrict__ wgt,
    const float* __restrict__ bias, float* __restrict__ out) {
  // LDS tile: [c4][y][x] with 4 channels packed per float4
  __shared__ v4f xs[4 * YR * XP];   // 56320 bytes

  const int tid = threadIdx.x;
  const int x0 = blockIdx.x * TILE_X;
  const int y0 = blockIdx.y * TILE_Y;
  const int b  = blockIdx.z;

  // Zero-fill only when the halo leaves the image (first/last tiles per dim)
  const bool border = (x0 == 0) || (y0 == 0) ||
                      (x0 + TILE_X == WW) || (y0 + TILE_Y == HH);
  if (border) {
    const v4f z = {0.f, 0.f, 0.f, 0.f};
    for (int t = tid; t < 4 * YR * XP; t += NTHR) xs[t] = z;
  }
  __syncthreads();

  // Stage x tile (incl. halo) into LDS via async copies.
  const float* xb = x + (size_t)b * CC * HH * WW;
  for (int t = tid; t < CC * YR * XR; t += NTHR) {
    int c  = t / (YR * XR);
    int r  = t - c * (YR * XR);
    int y  = r / XR;
    int xx = r - y * XR;
    int gy = y0 - RAD + y;
    int gx = x0 - RAD + xx;
    if ((unsigned)gy < (unsigned)HH && (unsigned)gx < (unsigned)WW) {
      float* lp = (float*)&xs[((c >> 2) * YR + y) * XP + xx] + (c & 3);
      async_g2l_b32(xb + ((size_t)c * HH + gy) * WW + gx, lp);
    }
  }
  wait_async0();
  __syncthreads();

  const int tx = tid & (TILE_X - 1);
  const int ty = tid >> 5;
  const int h  = y0 + ty;
  const int w  = x0 + tx;

  const v4f* bp = (const v4f*)bias;   // 16 floats, hipMalloc-aligned
  v4f a0 = bp[0], a1 = bp[1], a2 = bp[2], a3 = bp[3];

  // weight[b, k, h, w]; k-stride is uniform -> SGPR adds + vmem saddr form
  const float* wp = wgt + (((size_t)b * (KSZ * KSZ)) * HH + h) * WW + w;
  const v4f* r0 = &xs[(0 * YR + ty) * XP + tx];
  const v4f* r1 = &xs[(1 * YR + ty) * XP + tx];
  const v4f* r2 = &xs[(2 * YR + ty) * XP + tx];
  const v4f* r3 = &xs[(3 * YR + ty) * XP + tx];

#pragma unroll
  for (int i = 0; i < KSZ; ++i) {
#pragma unroll
    for (int j = 0; j < KSZ; ++j) {
      const float wv = wp[(size_t)(i * KSZ + j) * HH * WW];
      const int o = i * XP + j;       // constant ds offset immediate
      a0 += wv * r0[o];
      a1 += wv * r1[o];
      a2 += wv * r2[o];
      a3 += wv * r3[o];
    }
  }

  float* ob = out + (((size_t)b * CC) * HH + h) * WW + w;
#pragma unroll
  for (int ci = 0; ci < 4; ++ci) {
    ob[(size_t)(0 + ci) * HH * WW]  = a0[ci];
    ob[(size_t)(4 + ci) * HH * WW]  = a1[ci];
    ob[(size_t)(8 + ci) * HH * WW]  = a2[ci];
    ob[(size_t)(12 + ci) * HH * WW] = a3[ci];
  }
}

extern "C" void kernel_launch(void* const* d_in, const int* in_sizes, int n_in,
                              void* d_out, int out_size, void* d_ws, size_t ws_size,
                              hipStream_t stream) {
  (void)in_sizes; (void)n_in; (void)out_size; (void)d_ws; (void)ws_size;
  const float* x    = (const float*)d_in[0];
  const float* wgt  = (const float*)d_in[1];
  const float* bias = (const float*)d_in[2];
  float* out        = (float*)d_out;

  dim3 grid(WW / TILE_X, HH / TILE_Y, BB);   // (40, 45, 4)
  dim3 block(NTHR);
  wfilter7x7_kernel<<<grid, block, 0, stream>>>(x, wgt, bias, out);
}